// MSN_23424751633216
// MI455X (gfx1250) — compile-verified
//
#include <hip/hip_runtime.h>
#include <math.h>
#include <float.h>

// ---------------- types for WMMA ----------------
typedef __attribute__((ext_vector_type(2))) float v2f;
typedef __attribute__((ext_vector_type(8))) float v8f;

#define WMMA4(acc, a, b)                                                      \
  acc = __builtin_amdgcn_wmma_f32_16x16x4_f32(false, a, false, b, (short)0,   \
                                              acc, false, false)

// ---------------- problem constants ----------------
#define B_      4
#define NPART   2048
#define NPTS    8192
#define NPRIM   16
#define SPP     512
#define MDS_M   10240

// ---------------- workspace layout (float offsets) ----------------
static const size_t OFF_XENC  = 0;                       // 4*3*2048   = 24576
static const size_t OFF_GENC  = OFF_XENC  + 24576;       // 4096
static const size_t OFF_STYLE = OFF_GENC  + 4096;        // 4096
static const size_t OFF_XX    = OFF_STYLE + 4096;        // 4*4*10240  = 163840
static const size_t OFF_IDX   = OFF_XX    + 163840;      // 4*8192 ints= 32768
static const size_t OFF_XXG   = OFF_IDX   + 32768;       // 4*4*8192   = 131072
static const size_t OFF_GRES  = OFF_XXG   + 131072;      // 4096
static const size_t OFF_C4    = OFF_GRES  + 4096;        // 2048
static const size_t OFF_MEAN  = OFF_C4    + 2048;        // 4
static const size_t OFF_PEN   = OFF_MEAN  + 4;           // 1 (+3 pad)
static const size_t OFF_DELTA = OFF_PEN   + 4;           // 4*3*8192   = 98304
static const size_t OFF_H4    = OFF_DELTA + 98304;       // 4*3*512    = 6144
static const size_t OFF_BIG   = OFF_H4    + 6144;        // phase-overlaid region
// total = OFF_BIG + 13631488 floats  (~56.4 MB)

// ---------------- helpers ----------------
__device__ __forceinline__ void atomicMaxF(float* addr, float val) {
  unsigned int* ua = (unsigned int*)addr;
  unsigned int cur = __float_as_uint(*addr);
  while (__uint_as_float(cur) < val) {
    unsigned int prev = atomicCAS(ua, cur, __float_as_uint(val));
    if (prev == cur) break;
    cur = prev;
  }
}

// =====================================================================
// WMMA f32 GEMM core: computes a 16(M) x 64(N) strip per wave.
// All in-loop guards removed via loop-invariant address clamping;
// K-tail (C % 4 != 0) peeled as one masked iteration.
// =====================================================================
struct GemmAcc { v8f a0, a1, a2, a3; };

__device__ __forceinline__ GemmAcc wmma_strip(const float* __restrict__ W, int ldw,
                                              const float* __restrict__ IN, int ldin,
                                              int O, int C, int N,
                                              int mt, int nb, int lane) {
  const int l    = lane & 15;
  const int half = lane >> 4;
  const int row  = mt * 16 + l;
  const int rowA = (row < O) ? row : (O - 1);           // clamped, invariant
  int ncol = nb + l;
  ncol = (ncol < N) ? ncol : (N - 1);                   // clamped, invariant

  const float* Wp = W + (size_t)rowA * ldw + half * 2;  // a.x at +0, a.y at +1
  const float* B0 = IN + (size_t)(half * 2) * ldin + ncol;
  const float* B1 = B0 + ldin;
  const size_t bstep = (size_t)4 * ldin;

  GemmAcc r;
  r.a0 = (v8f){}; r.a1 = (v8f){}; r.a2 = (v8f){}; r.a3 = (v8f){};

  const int Cm = C & ~3;
  for (int k = 0; k < Cm; k += 4) {
    v2f a; a.x = Wp[0]; a.y = Wp[1];
    v2f b;
    b.x = B0[0];  b.y = B1[0];  WMMA4(r.a0, a, b);
    b.x = B0[16]; b.y = B1[16]; WMMA4(r.a1, a, b);
    b.x = B0[32]; b.y = B1[32]; WMMA4(r.a2, a, b);
    b.x = B0[48]; b.y = B1[48]; WMMA4(r.a3, a, b);
    Wp += 4; B0 += bstep; B1 += bstep;
  }
  if (Cm < C) {                                          // masked K tail (once)
    const int k0 = Cm + half * 2;
    const int k1 = k0 + 1;
    const int k0c = (k0 < C) ? k0 : (C - 1);
    const int k1c = (k1 < C) ? k1 : (C - 1);
    const float* Wr  = W + (size_t)rowA * ldw;
    const float* Br0 = IN + (size_t)k0c * ldin + ncol;
    const float* Br1 = IN + (size_t)k1c * ldin + ncol;
    float ax = Wr[k0c]; ax = (k0 < C) ? ax : 0.f;
    float ay = Wr[k1c]; ay = (k1 < C) ? ay : 0.f;
    v2f a; a.x = ax; a.y = ay;
    float bx, by; v2f b;
    bx = Br0[0];  by = Br1[0];
    b.x = (k0 < C) ? bx : 0.f; b.y = (k1 < C) ? by : 0.f; WMMA4(r.a0, a, b);
    bx = Br0[16]; by = Br1[16];
    b.x = (k0 < C) ? bx : 0.f; b.y = (k1 < C) ? by : 0.f; WMMA4(r.a1, a, b);
    bx = Br0[32]; by = Br1[32];
    b.x = (k0 < C) ? bx : 0.f; b.y = (k1 < C) ? by : 0.f; WMMA4(r.a2, a, b);
    bx = Br0[48]; by = Br1[48];
    b.x = (k0 < C) ? bx : 0.f; b.y = (k1 < C) ? by : 0.f; WMMA4(r.a3, a, b);
  }
  return r;
}

// OUT[o,n] = act( GEMM + bias[o] );  act: 0=none 1=relu 2=tanh
__global__ void k_wmma_gemm(const float* __restrict__ W, int ldw,
                            const float* __restrict__ bias, int biasBS,
                            const float* __restrict__ IN, int ldin, int inBS,
                            float* __restrict__ OUT, int ldout, int outBS,
                            int O, int C, int N, int act)
{
  const int b = blockIdx.y;
  IN  += (size_t)b * inBS;
  OUT += (size_t)b * outBS;
  const float* bp = bias + (size_t)b * biasBS;

  const int lane = threadIdx.x & 31;
  const int wid  = blockIdx.x * 8 + (threadIdx.x >> 5);
  const int Mt = (O + 15) >> 4;
  const int Ng = (N + 63) >> 6;
  if (wid >= Mt * Ng) return;                 // wave-uniform exit
  const int mt = wid % Mt;
  const int nb = (wid / Mt) * 64;

  GemmAcc r = wmma_strip(W, ldw, IN, ldin, O, C, N, mt, nb, lane);

  const int l = lane & 15, half = lane >> 4;
#define STORE_TILE(ACC, TOFF)                                                 \
  _Pragma("unroll")                                                           \
  for (int v = 0; v < 8; ++v) {                                               \
    const int rr = mt * 16 + v + half * 8;                                    \
    const int nn = nb + (TOFF) + l;                                           \
    if (rr < O && nn < N) {                                                   \
      float x = ACC[v] + bp[rr];                                              \
      if (act == 1) x = fmaxf(x, 0.f);                                        \
      else if (act == 2) x = tanhf(x);                                        \
      OUT[(size_t)rr * ldout + nn] = x;                                       \
    }                                                                         \
  }
  STORE_TILE(r.a0, 0)
  STORE_TILE(r.a1, 16)
  STORE_TILE(r.a2, 32)
  STORE_TILE(r.a3, 48)
#undef STORE_TILE
}

// GEMM fused with global max-pool over N: G[b*O+o] = max_n(conv+bias)
__global__ void k_wmma_gemm_maxpool(const float* __restrict__ W, int ldw,
                                    const float* __restrict__ bias,
                                    const float* __restrict__ IN, int ldin, int inBS,
                                    float* __restrict__ G,
                                    int O, int C, int N)
{
  const int b = blockIdx.y;
  IN += (size_t)b * inBS;
  float* Gb = G + (size_t)b * O;

  const int lane = threadIdx.x & 31;
  const int wid  = blockIdx.x * 8 + (threadIdx.x >> 5);
  const int Mt = (O + 15) >> 4;
  const int Ng = (N + 63) >> 6;
  if (wid >= Mt * Ng) return;
  const int mt = wid % Mt;
  const int nb = (wid / Mt) * 64;

  GemmAcc r = wmma_strip(W, ldw, IN, ldin, O, C, N, mt, nb, lane);

  const int l = lane & 15, half = lane >> 4;
#define MAX_TILE(ACC, TOFF)                                                   \
  _Pragma("unroll")                                                           \
  for (int v = 0; v < 8; ++v) {                                               \
    const int rr = mt * 16 + v + half * 8;                                    \
    const int nn = nb + (TOFF) + l;                                           \
    float val = (rr < O && nn < N) ? ACC[v] + bias[rr] : -FLT_MAX;            \
    val = fmaxf(val, __shfl_xor(val, 1, 32));                                 \
    val = fmaxf(val, __shfl_xor(val, 2, 32));                                 \
    val = fmaxf(val, __shfl_xor(val, 4, 32));                                 \
    val = fmaxf(val, __shfl_xor(val, 8, 32));                                 \
    if (l == 0 && rr < O) atomicMaxF(&Gb[rr], val);                           \
  }
  MAX_TILE(r.a0, 0)
  MAX_TILE(r.a1, 16)
  MAX_TILE(r.a2, 32)
  MAX_TILE(r.a3, 48)
#undef MAX_TILE
}

// =====================================================================
// small utility kernels
// =====================================================================
__global__ void k_init(float* genc, float* gres, float* meansum, float* pensum) {
  int t = blockIdx.x * blockDim.x + threadIdx.x;
  if (t < 4096) genc[t] = -FLT_MAX;
  else if (t < 8192) gres[t - 4096] = -FLT_MAX;
  if (t < 4) meansum[t] = 0.f;
  if (t == 4) pensum[0] = 0.f;
}

__global__ void k_transpose_pc(const float* __restrict__ pc, float* __restrict__ xenc) {
  int t = blockIdx.x * blockDim.x + threadIdx.x;
  if (t >= B_ * 3 * NPART) return;
  int n = t % NPART, c = (t / NPART) % 3, b = t / (3 * NPART);
  xenc[t] = pc[((size_t)b * NPART + n) * 3 + c];
}

__global__ void k_style(const float* __restrict__ g, const float* __restrict__ lw,
                        const float* __restrict__ lb, float* __restrict__ style) {
  int t = blockIdx.x * blockDim.x + threadIdx.x;
  if (t >= B_ * 1024) return;
  int b = t >> 10, o = t & 1023;
  const float* w = lw + (size_t)o * 1024;
  const float* gv = g + (size_t)b * 1024;
  float s = lb[o];
  for (int k = 0; k < 1024; ++k) s += w[k] * gv[k];
  style[t] = fmaxf(s, 0.f);
}

__global__ void k_build_y(const float* __restrict__ rgp, const float* __restrict__ style,
                          float* __restrict__ y) {
  int t = blockIdx.x * blockDim.x + threadIdx.x;
  if (t >= B_ * 1026 * SPP) return;
  int s = t % SPP, c = (t / SPP) % 1026, b = t / (SPP * 1026);
  y[t] = (c < 2) ? rgp[(b * 2 + c) * SPP + s] : style[b * 1024 + (c - 2)];
}

__global__ void k_scatter_coarse(const float* __restrict__ h4, float* __restrict__ coarse,
                                 float* __restrict__ xx, int p) {
  int t = blockIdx.x * blockDim.x + threadIdx.x;
  if (t >= B_ * SPP) return;
  int b = t >> 9, s = t & 511, n = p * SPP + s;
  float* xb = xx + (size_t)b * 4 * MDS_M;
#pragma unroll
  for (int k = 0; k < 3; ++k) {
    float v = h4[b * 1536 + k * SPP + s];
    coarse[((size_t)b * NPTS + n) * 3 + k] = v;
    xb[k * MDS_M + n] = v;
  }
  xb[3 * MDS_M + n] = 0.f;
}

__global__ void k_fill_tail(const float* __restrict__ xenc, float* __restrict__ xx) {
  int t = blockIdx.x * blockDim.x + threadIdx.x;
  if (t >= B_ * NPART) return;
  int b = t >> 11, m = t & 2047;
  float* xb = xx + (size_t)b * 4 * MDS_M;
  for (int c = 0; c < 3; ++c)
    xb[c * MDS_M + NPTS + m] = xenc[b * 3 * NPART + c * NPART + m];
  xb[3 * MDS_M + NPTS + m] = 1.f;
}

// =====================================================================
// MST expansion penalty: one 512-thread block per (b,prim) group.
// =====================================================================
__global__ void k_mst(const float* __restrict__ coarse, float* pen_sum, float* mean_sum) {
  const int grp = blockIdx.x;
  const int b = grp >> 4, p = grp & 15;
  const int tid = threadIdx.x;
  __shared__ float px[512], py[512], pz[512], mind[512], redv[512], edges[512];
  __shared__ int redi[512];
  __shared__ unsigned char vis[512];
  __shared__ int curj;
  __shared__ float meansh;

  const float* base = coarse + ((size_t)(b * NPTS + p * SPP)) * 3;
  px[tid] = base[tid * 3 + 0];
  py[tid] = base[tid * 3 + 1];
  pz[tid] = base[tid * 3 + 2];
  vis[tid] = 0;
  if (tid == 511) edges[511] = 0.f;
  __syncthreads();
  {
    float dx = px[tid] - px[0], dy = py[tid] - py[0], dz = pz[tid] - pz[0];
    mind[tid] = dx * dx + dy * dy + dz * dz;
  }
  if (tid == 0) vis[0] = 1;
  __syncthreads();

  for (int t = 0; t < SPP - 1; ++t) {
    redv[tid] = vis[tid] ? FLT_MAX : mind[tid];
    redi[tid] = tid;
    __syncthreads();
    for (int s = 256; s > 0; s >>= 1) {
      if (tid < s) {
        float v2 = redv[tid + s]; int i2 = redi[tid + s];
        if (v2 < redv[tid] || (v2 == redv[tid] && i2 < redi[tid])) {
          redv[tid] = v2; redi[tid] = i2;
        }
      }
      __syncthreads();
    }
    if (tid == 0) {
      int j = redi[0];
      curj = j; vis[j] = 1;
      edges[t] = sqrtf(redv[0]);
    }
    __syncthreads();
    int j = curj;
    float dx = px[tid] - px[j], dy = py[tid] - py[j], dz = pz[tid] - pz[j];
    float d2 = dx * dx + dy * dy + dz * dz;
    mind[tid] = fminf(mind[tid], d2);
    __syncthreads();
  }
  redv[tid] = (tid < 511) ? edges[tid] : 0.f;
  __syncthreads();
  for (int s = 256; s > 0; s >>= 1) { if (tid < s) redv[tid] += redv[tid + s]; __syncthreads(); }
  if (tid == 0) meansh = redv[0] / 511.f;
  __syncthreads();
  float mean = meansh;
  float pv = 0.f;
  if (tid < 511) { float e = edges[tid]; pv = (e > 1.5f * mean) ? e : 0.f; }
  redv[tid] = pv;
  __syncthreads();
  for (int s = 256; s > 0; s >>= 1) { if (tid < s) redv[tid] += redv[tid + s]; __syncthreads(); }
  if (tid == 0) {
    atomicAdd(pen_sum, redv[0]);
    atomicAdd(&mean_sum[b], mean);
  }
}

// =====================================================================
// Minimum-density sampling: one 1024-thread block per batch, persistent
// loop over 8191 sequential picks; 2 barriers/step.
// =====================================================================
__global__ void __launch_bounds__(1024)
k_mds(const float* __restrict__ xx, const float* __restrict__ mean_sum,
      int* __restrict__ idxout) {
  const int b = blockIdx.x;
  const int tid = threadIdx.x;
  const float* X = xx + (size_t)b * 4 * MDS_M;
  __shared__ unsigned int visbits[MDS_M / 32];
  __shared__ float wv[32];
  __shared__ int wi[32];
  __shared__ float pcur[3];

  if (tid < MDS_M / 32) visbits[tid] = 0u;
  float sigma = mean_sum[b] * (1.f / 16.f);
  float inv = 1.f / (2.f * sigma * sigma);

  float ptx[10], pty[10], ptz[10], dens[10];
#pragma unroll
  for (int k = 0; k < 10; ++k) {
    int m = k * 1024 + tid;
    ptx[k] = X[m];
    pty[k] = X[MDS_M + m];
    ptz[k] = X[2 * MDS_M + m];
    dens[k] = 0.f;
  }
  __syncthreads();
  if (tid == 0) {
    visbits[0] = 1u;
    idxout[b * NPTS] = 0;
    pcur[0] = X[0]; pcur[1] = X[MDS_M]; pcur[2] = X[2 * MDS_M];
  }
  __syncthreads();

  for (int t = 1; t < NPTS; ++t) {
    const float cx = pcur[0], cy = pcur[1], cz = pcur[2];
    float bestv = FLT_MAX;
    int besti = MDS_M;
#pragma unroll
    for (int k = 0; k < 10; ++k) {
      int m = k * 1024 + tid;
      float dx = ptx[k] - cx, dy = pty[k] - cy, dz = ptz[k] - cz;
      float d2 = dx * dx + dy * dy + dz * dz;
      dens[k] += __expf(-d2 * inv);
      bool visited = (visbits[m >> 5] >> (m & 31)) & 1u;
      float v = visited ? 1e10f : dens[k];
      if (v < bestv || (v == bestv && m < besti)) { bestv = v; besti = m; }
    }
    for (int off = 16; off >= 1; off >>= 1) {
      float ov = __shfl_xor(bestv, off, 32);
      int oi = __shfl_xor(besti, off, 32);
      if (ov < bestv || (ov == bestv && oi < besti)) { bestv = ov; besti = oi; }
    }
    if ((tid & 31) == 0) { wv[tid >> 5] = bestv; wi[tid >> 5] = besti; }
    __syncthreads();
    if (tid < 32) {
      bestv = wv[tid]; besti = wi[tid];
      for (int off = 16; off >= 1; off >>= 1) {
        float ov = __shfl_xor(bestv, off, 32);
        int oi = __shfl_xor(besti, off, 32);
        if (ov < bestv || (ov == bestv && oi < besti)) { bestv = ov; besti = oi; }
      }
      if (tid == 0) {
        int j = besti;
        visbits[j >> 5] |= (1u << (j & 31));
        idxout[b * NPTS + t] = j;
        pcur[0] = X[j]; pcur[1] = X[MDS_M + j]; pcur[2] = X[2 * MDS_M + j];
      }
    }
    __syncthreads();
  }
}

__global__ void k_gather(const float* __restrict__ xx, const int* __restrict__ idx,
                         float* __restrict__ xxg) {
  int t = blockIdx.x * blockDim.x + threadIdx.x;
  if (t >= B_ * 4 * NPTS) return;
  int n = t % NPTS, c = (t / NPTS) % 4, b = t / (4 * NPTS);
  int j = idx[b * NPTS + n];
  xxg[t] = xx[(size_t)b * 4 * MDS_M + c * MDS_M + j];
}

__global__ void k_c4(const float* __restrict__ w4, const float* __restrict__ b4,
                     const float* __restrict__ g, float* __restrict__ c4) {
  int t = blockIdx.x * blockDim.x + threadIdx.x;
  if (t >= B_ * 512) return;
  int b = t >> 9, o = t & 511;
  const float* w = w4 + (size_t)o * 1088 + 64;
  const float* gv = g + (size_t)b * 1024;
  float s = b4[o];
  for (int k = 0; k < 1024; ++k) s += w[k] * gv[k];
  c4[t] = s;
}

__global__ void k_refine(const float* __restrict__ xxg, const float* __restrict__ delta,
                         float* __restrict__ out_refine) {
  int t = blockIdx.x * blockDim.x + threadIdx.x;
  if (t >= B_ * NPTS * 3) return;
  int k = t % 3, n = (t / 3) % NPTS, b = t / (3 * NPTS);
  out_refine[t] = xxg[(size_t)b * 4 * NPTS + k * NPTS + n] +
                  delta[(size_t)b * 3 * NPTS + k * NPTS + n];
}

__global__ void k_loss(const float* pen_sum, float* out_loss) {
  if (threadIdx.x == 0 && blockIdx.x == 0)
    out_loss[0] = pen_sum[0] / (float)(B_ * NPTS);
}

// =====================================================================
// host-side launcher
// =====================================================================
static inline void gemm(const float* W, int ldw, const float* bias, int biasBS,
                        const float* IN, int ldin, int inBS,
                        float* OUT, int ldout, int outBS,
                        int O, int C, int N, int act, hipStream_t s) {
  int Mt = (O + 15) / 16, Ng = (N + 63) / 64;
  int blocks = (Mt * Ng + 7) / 8;
  dim3 g(blocks, B_, 1);
  k_wmma_gemm<<<g, 256, 0, s>>>(W, ldw, bias, biasBS, IN, ldin, inBS,
                                OUT, ldout, outBS, O, C, N, act);
}

static inline void gemm_max(const float* W, int ldw, const float* bias,
                            const float* IN, int ldin, int inBS, float* G,
                            int O, int C, int N, hipStream_t s) {
  int Mt = (O + 15) / 16, Ng = (N + 63) / 64;
  int blocks = (Mt * Ng + 7) / 8;
  dim3 g(blocks, B_, 1);
  k_wmma_gemm_maxpool<<<g, 256, 0, s>>>(W, ldw, bias, IN, ldin, inBS, G, O, C, N);
}

extern "C" void kernel_launch(void* const* d_in, const int* in_sizes, int n_in,
                              void* d_out, int out_size, void* d_ws, size_t ws_size,
                              hipStream_t stream) {
  (void)in_sizes; (void)n_in; (void)out_size; (void)ws_size;
  const float* pc  = (const float*)d_in[0];
  const float* rg  = (const float*)d_in[1];
  const float* ew1 = (const float*)d_in[2];  const float* eb1 = (const float*)d_in[3];
  const float* ew2 = (const float*)d_in[4];  const float* eb2 = (const float*)d_in[5];
  const float* ew3 = (const float*)d_in[6];  const float* eb3 = (const float*)d_in[7];
  const float* elw = (const float*)d_in[8];  const float* elb = (const float*)d_in[9];
  const float* dw1 = (const float*)d_in[10]; const float* db1 = (const float*)d_in[11];
  const float* dw2 = (const float*)d_in[12]; const float* db2 = (const float*)d_in[13];
  const float* dw3 = (const float*)d_in[14]; const float* db3 = (const float*)d_in[15];
  const float* dw4 = (const float*)d_in[16]; const float* db4 = (const float*)d_in[17];
  const float* rw1 = (const float*)d_in[18]; const float* rb1 = (const float*)d_in[19];
  const float* rw2 = (const float*)d_in[20]; const float* rb2 = (const float*)d_in[21];
  const float* rw3 = (const float*)d_in[22]; const float* rb3 = (const float*)d_in[23];
  const float* rw4 = (const float*)d_in[24]; const float* rb4 = (const float*)d_in[25];
  const float* rw5 = (const float*)d_in[26]; const float* rb5 = (const float*)d_in[27];
  const float* rw6 = (const float*)d_in[28]; const float* rb6 = (const float*)d_in[29];
  const float* rw7 = (const float*)d_in[30]; const float* rb7 = (const float*)d_in[31];

  float* out = (float*)d_out;                  // coarse | refine | loss
  float* ws  = (float*)d_ws;
  float* xenc  = ws + OFF_XENC;
  float* genc  = ws + OFF_GENC;
  float* style = ws + OFF_STYLE;
  float* xx    = ws + OFF_XX;
  int*   idxb  = (int*)(ws + OFF_IDX);
  float* xxg   = ws + OFF_XXG;
  float* gres  = ws + OFF_GRES;
  float* c4    = ws + OFF_C4;
  float* msum  = ws + OFF_MEAN;
  float* psum  = ws + OFF_PEN;
  float* delta = ws + OFF_DELTA;
  float* h4    = ws + OFF_H4;
  float* big   = ws + OFF_BIG;

  // ---- init accumulators ----
  k_init<<<32, 256, 0, stream>>>(genc, gres, msum, psum);

  // ---- encoder ----
  k_transpose_pc<<<(B_ * 3 * NPART + 255) / 256, 256, 0, stream>>>(pc, xenc);
  float* h1 = big;                 // [B,128,2048]
  float* h2 = big + 1048576;       // [B,256,2048]
  gemm(ew1, 3, eb1, 0, xenc, NPART, 3 * NPART, h1, NPART, 128 * NPART,
       128, 3, NPART, 1, stream);
  gemm(ew2, 128, eb2, 0, h1, NPART, 128 * NPART, h2, NPART, 256 * NPART,
       256, 128, NPART, 1, stream);
  gemm_max(ew3, 256, eb3, h2, NPART, 256 * NPART, genc, 1024, 256, NPART, stream);
  k_style<<<(B_ * 1024 + 255) / 256, 256, 0, stream>>>(genc, elw, elb, style);

  // ---- decoder: loop over primitives, reusing buffers ----
  float* y   = big;                 // [B,1026,512]
  float* dh1 = big + 2101248;       // [B,1026,512]
  float* dh2 = big + 4202496;       // [B,513,512]
  float* dh3 = big + 5253120;       // [B,256,512]
  for (int p = 0; p < NPRIM; ++p) {
    k_build_y<<<(B_ * 1026 * SPP + 255) / 256, 256, 0, stream>>>(
        rg + (size_t)p * B_ * 2 * SPP, style, y);
    gemm(dw1 + (size_t)p * 1026 * 1026, 1026, db1 + p * 1026, 0,
         y, SPP, 1026 * SPP, dh1, SPP, 1026 * SPP, 1026, 1026, SPP, 1, stream);
    gemm(dw2 + (size_t)p * 513 * 1026, 1026, db2 + p * 513, 0,
         dh1, SPP, 1026 * SPP, dh2, SPP, 513 * SPP, 513, 1026, SPP, 1, stream);
    gemm(dw3 + (size_t)p * 256 * 513, 513, db3 + p * 256, 0,
         dh2, SPP, 513 * SPP, dh3, SPP, 256 * SPP, 256, 513, SPP, 1, stream);
    gemm(dw4 + (size_t)p * 3 * 256, 256, db4 + p * 3, 0,
         dh3, SPP, 256 * SPP, h4, SPP, 3 * SPP, 3, 256, SPP, 2, stream);
    k_scatter_coarse<<<(B_ * SPP + 255) / 256, 256, 0, stream>>>(h4, out, xx, p);
  }
  k_fill_tail<<<(B_ * NPART + 255) / 256, 256, 0, stream>>>(xenc, xx);

  // ---- expansion penalty (MST) ----
  k_mst<<<B_ * NPRIM, 512, 0, stream>>>(out, psum, msum);

  // ---- minimum-density sampling ----
  k_mds<<<B_, 1024, 0, stream>>>(xx, msum, idxb);
  k_gather<<<(B_ * 4 * NPTS + 255) / 256, 256, 0, stream>>>(xx, idxb, xxg);

  // ---- residual network ----
  float* pf   = big;                 // [B,64,8192]
  float* rh2  = big + 2097152;       // [B,128,8192]
  float* rh4c = big + 6291456;       // [B,512,2048]
  float* rh5c = big + 10485760;      // [B,256,2048]
  float* rh6c = big + 12582912;      // [B,128,2048]
  gemm(rw1, 4, rb1, 0, xxg, NPTS, 4 * NPTS, pf, NPTS, 64 * NPTS,
       64, 4, NPTS, 1, stream);
  gemm(rw2, 64, rb2, 0, pf, NPTS, 64 * NPTS, rh2, NPTS, 128 * NPTS,
       128, 64, NPTS, 1, stream);
  gemm_max(rw3, 128, rb3, rh2, NPTS, 128 * NPTS, gres, 1024, 128, NPTS, stream);
  k_c4<<<(B_ * 512 + 255) / 256, 256, 0, stream>>>(rw4, rb4, gres, c4);

  const int CHUNK = 2048;
  for (int n0 = 0; n0 < NPTS; n0 += CHUNK) {
    gemm(rw4, 1088, c4, 512, pf + n0, NPTS, 64 * NPTS,
         rh4c, CHUNK, 512 * CHUNK, 512, 64, CHUNK, 1, stream);
    gemm(rw5, 512, rb5, 0, rh4c, CHUNK, 512 * CHUNK,
         rh5c, CHUNK, 256 * CHUNK, 256, 512, CHUNK, 1, stream);
    gemm(rw6, 256, rb6, 0, rh5c, CHUNK, 256 * CHUNK,
         rh6c, CHUNK, 128 * CHUNK, 128, 256, CHUNK, 1, stream);
    gemm(rw7, 128, rb7, 0, rh6c, CHUNK, 128 * CHUNK,
         delta + n0, NPTS, 3 * NPTS, 3, 128, CHUNK, 2, stream);
  }
  k_refine<<<(B_ * NPTS * 3 + 255) / 256, 256, 0, stream>>>(xxg, delta,
                                                            out + (size_t)B_ * NPTS * 3);
  k_loss<<<1, 1, 0, stream>>>(psum, out + (size_t)2 * B_ * NPTS * 3);
}